// SSD_39779987095892
// MI455X (gfx1250) — compile-verified
//
#include <hip/hip_runtime.h>
#include <hip/hip_bf16.h>
#include <math.h>
#include <stdint.h>

// ---- problem constants (match reference) ----
#define BATCH   32
#define P_NUM   8732
#define NCLS    21
#define FG      20
#define TOPK    200
#define CONF_TH 0.01f
#define NMS_TH  0.45f
#define EPS_V   1e-6f
#define THREADS 256
#define NWAVES  (THREADS / 32)
#define CHUNK   16
#define NCHUNK  ((P_NUM + CHUNK - 1) / CHUNK)   // 546

__device__ __forceinline__ float clampf(float v, float lo, float hi) {
    return fminf(fmaxf(v, lo), hi);
}

// ---------------------------------------------------------------------------
// Kernel 1: decode boxes + clipped softmax + validity masking.
// One thread per (batch, prior). Writes:
//   boxes   [B, P, 4]           (x1,y1,x2,y2 clamped to [0,1])
//   mscores [B, FG, P]          score if (score>thr && box valid) else -1.0
// Class-major score layout => kernel 2 streams one contiguous 34.9KB row.
// ---------------------------------------------------------------------------
__global__ void ssd_decode_softmax(const float* __restrict__ loc,
                                   const float* __restrict__ conf,
                                   const float* __restrict__ priors,
                                   float* __restrict__ boxes,
                                   float* __restrict__ mscores)
{
    int t = blockIdx.x * blockDim.x + threadIdx.x;
    if (t >= BATCH * P_NUM) return;
    int b = t / P_NUM;
    int p = t - b * P_NUM;

    // speculative prefetch of a future conf row (emits global_prefetch_b8)
    if (t + THREADS < BATCH * P_NUM)
        __builtin_prefetch(conf + (size_t)(t + THREADS) * NCLS, 0, 0);

    // ---- decode (calc_predicted_bboxes) ----
    float pr0 = fmaxf(priors[p * 4 + 0], EPS_V);
    float pr1 = fmaxf(priors[p * 4 + 1], EPS_V);
    float pr2 = fmaxf(priors[p * 4 + 2], EPS_V);
    float pr3 = fmaxf(priors[p * 4 + 3], EPS_V);

    const float* lv = loc + (size_t)t * 4;
    float l0 = lv[0], l1 = lv[1], l2 = lv[2], l3 = lv[3];

    float cx = pr0 + l0 * 0.1f * pr2;
    float cy = pr1 + l1 * 0.1f * pr3;
    float lw = clampf(l2 * 0.2f, -4.0f, 4.0f);
    float lh = clampf(l3 * 0.2f, -4.0f, 4.0f);
    float w  = pr2 * expf(lw);
    float h  = pr3 * expf(lh);

    float x1 = clampf(cx - 0.5f * w, 0.0f, 1.0f);
    float y1 = clampf(cy - 0.5f * h, 0.0f, 1.0f);
    float x2 = clampf(cx + 0.5f * w, 0.0f, 1.0f);
    float y2 = clampf(cy + 0.5f * h, 0.0f, 1.0f);

    float* bx = boxes + (size_t)t * 4;
    bx[0] = x1; bx[1] = y1; bx[2] = x2; bx[3] = y2;

    bool box_ok = (x2 > x1 + EPS_V) && (y2 > y1 + EPS_V);

    // ---- clipped softmax over 21 classes ----
    const float* cv = conf + (size_t)t * NCLS;
    float x[NCLS];
    float m = -1e30f;
    #pragma unroll
    for (int c = 0; c < NCLS; ++c) {
        x[c] = clampf(cv[c], -100.0f, 100.0f);
        m = fmaxf(m, x[c]);
    }
    float s = 0.0f;
    #pragma unroll
    for (int c = 0; c < NCLS; ++c) {
        x[c] = expf(x[c] - m);
        s += x[c];
    }
    float inv = 1.0f / s;
    #pragma unroll
    for (int c = 1; c < NCLS; ++c) {
        float prob = x[c] * inv;
        float msc = (box_ok && prob > CONF_TH) ? prob : -1.0f;
        mscores[((size_t)b * FG + (c - 1)) * P_NUM + p] = msc;
    }
}

// ---------------------------------------------------------------------------
// Kernel 2: per-(image,class) top-K + NMS + stable compaction.
// One 256-thread workgroup per (b, class). All working state in LDS (~48KB,
// well under the 320KB WGP budget). Score row is staged LDS-side via gfx1250
// async global->LDS copies (ASYNCcnt-tracked).
// ---------------------------------------------------------------------------
__global__ void ssd_nms(const float* __restrict__ boxes,
                        const float* __restrict__ mscores,
                        float* __restrict__ out)
{
    __shared__ __align__(16) float s_scores[P_NUM];
    __shared__ float s_cmax[NCHUNK];    // per-16-element chunk max value
    __shared__ int   s_cidx[NCHUNK];    // per-chunk argmax (element index)
    __shared__ float s_rv[NWAVES];
    __shared__ int   s_ri[NWAVES];
    __shared__ float s_top_s[TOPK];
    __shared__ int   s_top_i[TOPK];
    __shared__ float s_box[TOPK * 4];
    __shared__ float s_area[TOPK];
    __shared__ int   s_alive[TOPK];
    __shared__ int   s_inv[TOPK];
    __shared__ int   s_cnt;

    const int tid = threadIdx.x;
    const int bc  = blockIdx.x;        // b * NCLS + cls
    const int b   = bc / NCLS;
    const int cls = bc - b * NCLS;

    float* orow = out + (size_t)bc * TOPK * 5;

    // Background class: output slab is all zeros.
    if (cls == 0) {
        for (int i = tid; i < TOPK * 5; i += THREADS) orow[i] = 0.0f;
        return;
    }

    // ---- Phase A: async-copy the masked score row into LDS ----
    const float* srow = mscores + ((size_t)b * FG + (cls - 1)) * P_NUM;
    unsigned lds_base = (unsigned)(uintptr_t)(&s_scores[0]);
    // P_NUM = 8732 floats = exactly 2183 x 16B transfers
    for (int c = tid; c < (P_NUM / 4); c += THREADS) {
        unsigned laddr = lds_base + 16u * (unsigned)c;
        unsigned long long gaddr = (unsigned long long)(uintptr_t)(srow + 4 * c);
        asm volatile("global_load_async_to_lds_b128 %0, %1, off"
                     :: "v"(laddr), "v"(gaddr) : "memory");
    }
#if __has_builtin(__builtin_amdgcn_s_wait_asynccnt)
    __builtin_amdgcn_s_wait_asynccnt(0);
#else
    asm volatile("s_wait_asynccnt 0" ::: "memory");
#endif
    __syncthreads();

    // ---- Phase B0: build per-chunk maxima (tie -> lowest index) ----
    for (int c = tid; c < NCHUNK; c += THREADS) {
        int s0 = c * CHUNK;
        int e  = min(s0 + CHUNK, P_NUM);
        float bv = -3.0e38f;
        int   bi = s0;
        for (int p = s0; p < e; ++p) {
            float v = s_scores[p];
            if (v > bv) { bv = v; bi = p; }   // ascending p => first max wins
        }
        s_cmax[c] = bv;
        s_cidx[c] = bi;
    }
    __syncthreads();

    // ---- Phase B: top-200 via chunk-max tournament ----
    // Tie rule (equal score -> lower element index) matches jax.lax.top_k.
    for (int k = 0; k < TOPK; ++k) {
        float best = -3.0e38f;
        int   bi   = 0x7fffffff;
        for (int c = tid; c < NCHUNK; c += THREADS) {
            float v = s_cmax[c];
            int   i2 = s_cidx[c];
            if (v > best || (v == best && i2 < bi)) { best = v; bi = i2; }
        }
        // wave32 reduction, no barrier
        #pragma unroll
        for (int off = 16; off > 0; off >>= 1) {
            float ov = __shfl_down(best, off);
            int   oi = __shfl_down(bi, off);
            if (ov > best || (ov == best && oi < bi)) { best = ov; bi = oi; }
        }
        int wid = tid >> 5, lane = tid & 31;
        if (lane == 0) { s_rv[wid] = best; s_ri[wid] = bi; }
        __syncthreads();
        if (tid == 0) {
            float bv = s_rv[0];
            int bidx = s_ri[0];
            #pragma unroll
            for (int w2 = 1; w2 < NWAVES; ++w2) {
                float ov = s_rv[w2];
                int   oi = s_ri[w2];
                if (ov > bv || (ov == bv && oi < bidx)) { bv = ov; bidx = oi; }
            }
            s_top_s[k] = bv;
            s_top_i[k] = bidx;
            s_scores[bidx] = -3.0e38f;       // remove winner
            // recompute only the winner's chunk
            int c  = bidx >> 4;
            int s0 = c << 4;
            int e  = min(s0 + CHUNK, P_NUM);
            float nbv = -3.0e38f;
            int   nbi = s0;
            for (int p = s0; p < e; ++p) {
                float v = s_scores[p];
                if (v > nbv) { nbv = v; nbi = p; }
            }
            s_cmax[c] = nbv;
            s_cidx[c] = nbi;
        }
        __syncthreads();
    }

    // ---- Phase C: gather candidate boxes, areas, init alive ----
    for (int k = tid; k < TOPK; k += THREADS) {
        int idx = s_top_i[k];
        float4 v = ((const float4*)boxes)[(size_t)b * P_NUM + idx];  // 16B/box
        s_box[k * 4 + 0] = v.x;
        s_box[k * 4 + 1] = v.y;
        s_box[k * 4 + 2] = v.z;
        s_box[k * 4 + 3] = v.w;
        s_area[k] = (v.z - v.x) * (v.w - v.y);
        s_alive[k] = 1;
    }

    // ---- Phase D: serial NMS (mirrors reference fori_loop) ----
    for (int i = 0; i < TOPK; ++i) {
        __syncthreads();
        bool keep_i = (s_alive[i] != 0) && (s_top_s[i] > CONF_TH); // uniform
        if (!keep_i) continue;
        float ix1 = s_box[i * 4 + 0], iy1 = s_box[i * 4 + 1];
        float ix2 = s_box[i * 4 + 2], iy2 = s_box[i * 4 + 3];
        float ia  = s_area[i];
        for (int j = i + 1 + tid; j < TOPK; j += THREADS) {
            if (!s_alive[j]) continue;
            float lx = fmaxf(ix1, s_box[j * 4 + 0]);
            float ly = fmaxf(iy1, s_box[j * 4 + 1]);
            float rx = fminf(ix2, s_box[j * 4 + 2]);
            float ry = fminf(iy2, s_box[j * 4 + 3]);
            float iw = fmaxf(rx - lx, 0.0f);
            float ih = fmaxf(ry - ly, 0.0f);
            float inter = iw * ih;
            float uni = ia + s_area[j] - inter;
            float iou = inter / fmaxf(uni, 1e-12f);
            if (iou > NMS_TH) s_alive[j] = 0;
        }
    }
    __syncthreads();

    // ---- Phase E: stable compaction (kept rows front, rest zero) ----
    if (tid == 0) {
        int cnt = 0;
        for (int k = 0; k < TOPK; ++k) {
            if (s_alive[k] && s_top_s[k] > CONF_TH) s_inv[cnt++] = k;
        }
        s_cnt = cnt;
    }
    __syncthreads();
    int cnt = s_cnt;
    for (int s = tid; s < TOPK; s += THREADS) {
        float r0 = 0.f, r1 = 0.f, r2 = 0.f, r3 = 0.f, r4 = 0.f;
        if (s < cnt) {
            int k = s_inv[s];
            r0 = s_top_s[k];
            r1 = s_box[k * 4 + 0];
            r2 = s_box[k * 4 + 1];
            r3 = s_box[k * 4 + 2];
            r4 = s_box[k * 4 + 3];
        }
        float* r = orow + (size_t)s * 5;
        r[0] = r0; r[1] = r1; r[2] = r2; r[3] = r3; r[4] = r4;
    }
}

// ---------------------------------------------------------------------------
extern "C" void kernel_launch(void* const* d_in, const int* in_sizes, int n_in,
                              void* d_out, int out_size, void* d_ws, size_t ws_size,
                              hipStream_t stream) {
    (void)in_sizes; (void)n_in; (void)out_size; (void)ws_size;
    const float* loc    = (const float*)d_in[0];  // [B, P, 4]
    const float* conf   = (const float*)d_in[1];  // [B, P, 21]
    const float* priors = (const float*)d_in[2];  // [P, 4]
    float*       out    = (float*)d_out;          // [B, 21, 200, 5]

    // workspace partition: boxes [B,P,4] then masked scores [B,FG,P]
    float* boxes   = (float*)d_ws;
    float* mscores = boxes + (size_t)BATCH * P_NUM * 4;

    int n = BATCH * P_NUM;
    ssd_decode_softmax<<<(n + THREADS - 1) / THREADS, THREADS, 0, stream>>>(
        loc, conf, priors, boxes, mscores);

    ssd_nms<<<BATCH * NCLS, THREADS, 0, stream>>>(boxes, mscores, out);
}